// DGCNN_lean_1417339207992
// MI455X (gfx1250) — compile-verified
//
#include <hip/hip_runtime.h>
#include <hip/hip_bf16.h>

#define NPTS   8192
#define KNBR   20
#define MEDGE  (NPTS * KNBR)      /* 163840 edge rows */
#define CONVBLK (MEDGE / 32)      /* 5120 blocks, 32 rows each */
#define BN_EPS 1e-5f

typedef __attribute__((ext_vector_type(2))) float v2f;
typedef __attribute__((ext_vector_type(8))) float v8f;

__device__ __forceinline__ v8f wmma4_f32(v2f a, v2f b, v8f c) {
    // D = A(16x4) * B(4x16) + C(16x16), all f32, wave32
    return __builtin_amdgcn_wmma_f32_16x16x4_f32(false, a, false, b, (short)0, c, false, false);
}

__device__ __forceinline__ float lrelu(float v) { return v >= 0.f ? v : 0.2f * v; }

// ---------------------------------------------------------------------------
// KNN: one block per query point i. pd computed exactly like the reference:
// pd = -xx_i - (-2*dot) - xx_j. 20 rounds of argmax with tie -> smaller index.
// ---------------------------------------------------------------------------
__global__ void __launch_bounds__(256) knn_kernel(const float* __restrict__ x,
                                                  int* __restrict__ idxout)
{
    __shared__ float pd[NPTS];
    __shared__ float bv[256];
    __shared__ int   bj[256];
    const int i = blockIdx.x;
    const int t = threadIdx.x;
    const float xi = x[64 * NPTS + i];
    const float yi = x[65 * NPTS + i];
    const float zi = x[66 * NPTS + i];
    const float xxi = xi * xi + yi * yi + zi * zi;
    for (int j = t; j < NPTS; j += 256) {
        float xj = x[64 * NPTS + j];
        float yj = x[65 * NPTS + j];
        float zj = x[66 * NPTS + j];
        float dot = xi * xj + yi * yj + zi * zj;
        float xxj = xj * xj + yj * yj + zj * zj;
        pd[j] = -xxi - (-2.f * dot) - xxj;
    }
    __syncthreads();
    const float NEGINF = -__builtin_inff();
    for (int r = 0; r < KNBR; ++r) {
        float best = NEGINF; int bidx = 0x7fffffff;
        for (int j = t; j < NPTS; j += 256) {
            float v = pd[j];
            if (v > best) { best = v; bidx = j; }   // ascending j + strict > => smallest idx on tie
        }
        bv[t] = best; bj[t] = bidx;
        __syncthreads();
        for (int s = 128; s > 0; s >>= 1) {
            if (t < s) {
                float v2 = bv[t + s]; int j2 = bj[t + s];
                if (v2 > bv[t] || (v2 == bv[t] && j2 < bj[t])) { bv[t] = v2; bj[t] = j2; }
            }
            __syncthreads();
        }
        if (t == 0) { idxout[i * KNBR + r] = bj[0]; pd[bj[0]] = NEGINF; }
        __syncthreads();
    }
}

// ---------------------------------------------------------------------------
// Shared conv epilogue: store raw y tile + per-block per-channel sum/sumsq.
// Block = 32 rows x 64 channels; 8 waves, wave w -> tile (mt=w/4, nt=w%4).
// ---------------------------------------------------------------------------
__device__ __forceinline__ void conv_store_stats(
    v8f acc, float* __restrict__ y, float* __restrict__ pS, float* __restrict__ pQ,
    int e0, int mt, int nt, int lhalf, int l16, int tid, int blk)
{
    __shared__ float redS[256];
    __shared__ float redQ[256];
    const int col = nt * 16 + l16;
    float s1 = 0.f, s2 = 0.f;
#pragma unroll
    for (int v = 0; v < 8; ++v) {
        float val = acc[v];
        int m = mt * 16 + lhalf * 8 + v;                 // D layout: vgpr v -> row v + 8*lhalf
        y[(size_t)(e0 + m) * 64 + col] = val;
        s1 += val;
        s2 += val * val;
    }
    redS[tid] = s1; redQ[tid] = s2;
    __syncthreads();
    if (tid < 64) {
        int ch = tid, nt2 = ch >> 4, l = ch & 15;
        int i0 = nt2 * 32 + l, i1 = nt2 * 32 + 16 + l;
        int i2 = (nt2 + 4) * 32 + l, i3 = (nt2 + 4) * 32 + 16 + l;
        pS[(size_t)blk * 64 + ch] = redS[i0] + redS[i1] + redS[i2] + redS[i3];
        pQ[(size_t)blk * 64 + ch] = redQ[i0] + redQ[i1] + redQ[i2] + redQ[i3];
    }
}

// ---------------------------------------------------------------------------
// conv1: edge features from x (channel-major, 67 ch), Cin = 134 padded to 136.
// A and B tiles staged in LDS; inner loop is branch-free ds_load + wmma.
// ---------------------------------------------------------------------------
__global__ void __launch_bounds__(256) conv1_kernel(
    const float* __restrict__ x, const int* __restrict__ idx,
    const float* __restrict__ W, float* __restrict__ y,
    float* __restrict__ pS, float* __restrict__ pQ)
{
    __shared__ float As[32 * 137];    // pitch 137 (bank stride 9 -> conflict-free cols)
    __shared__ float Bs[136 * 65];    // Bs[c*65 + o], pitch 65 (bank stride 1)
    const int tid = threadIdx.x, lane = tid & 31, wave = tid >> 5;
    const int mt = wave >> 2, nt = wave & 3, lhalf = lane >> 4, l16 = lane & 15;
    const int blk = blockIdx.x, e0 = blk * 32;

    // stage A: thread group of 8 per row
    {
        const int tr = tid >> 3;             // 0..31
        const int e = e0 + tr;
        const int n = e / KNBR;
        const int j = idx[e];
        for (int c = (tid & 7); c < 136; c += 8) {
            float f;
            if (c < 67)       f = x[c * NPTS + j] - x[c * NPTS + n];
            else if (c < 134) f = x[(c - 67) * NPTS + n];
            else              f = 0.f;
            As[tr * 137 + c] = f;
        }
    }
    // stage B: coalesced read of W (64 x 134), scatter to Bs[c][o]
    for (int e = tid; e < 64 * 136; e += 256) {
        int o = e / 136, c = e % 136;
        Bs[c * 65 + o] = (c < 134) ? W[o * 134 + c] : 0.f;
    }
    __syncthreads();

    const int arow = (mt * 16 + l16) * 137;
    const int col = nt * 16 + l16;
    const int koff = lhalf * 2;
    v8f acc = {};
#pragma unroll
    for (int kb = 0; kb < 136; kb += 4) {
        v2f a, b;
        a.x = As[arow + kb + koff];
        a.y = As[arow + kb + koff + 1];
        b.x = Bs[(kb + koff) * 65 + col];
        b.y = Bs[(kb + koff + 1) * 65 + col];
        acc = wmma4_f32(a, b, acc);
    }
    conv_store_stats(acc, y, pS, pQ, e0, mt, nt, lhalf, l16, tid, blk);
}

// ---------------------------------------------------------------------------
// conv3/conv5: edge features from F (N x 64 row-major), Cin = 128.
// ---------------------------------------------------------------------------
__global__ void __launch_bounds__(256) conv_edge64_kernel(
    const float* __restrict__ F, const int* __restrict__ idx,
    const float* __restrict__ W, float* __restrict__ y,
    float* __restrict__ pS, float* __restrict__ pQ)
{
    __shared__ float As[32 * 129];
    __shared__ float Bs[128 * 65];
    const int tid = threadIdx.x, lane = tid & 31, wave = tid >> 5;
    const int mt = wave >> 2, nt = wave & 3, lhalf = lane >> 4, l16 = lane & 15;
    const int blk = blockIdx.x, e0 = blk * 32;

    {
        const int tr = tid >> 3;
        const int e = e0 + tr;
        const int n = e / KNBR;
        const int j = idx[e];
        for (int c = (tid & 7); c < 128; c += 8) {
            float f = (c < 64) ? (F[(size_t)j * 64 + c] - F[(size_t)n * 64 + c])
                               : F[(size_t)n * 64 + c - 64];
            As[tr * 129 + c] = f;
        }
    }
    for (int e = tid; e < 64 * 128; e += 256) {
        int o = e >> 7, c = e & 127;
        Bs[c * 65 + o] = W[o * 128 + c];
    }
    __syncthreads();

    const int arow = (mt * 16 + l16) * 129;
    const int col = nt * 16 + l16;
    const int koff = lhalf * 2;
    v8f acc = {};
#pragma unroll
    for (int kb = 0; kb < 128; kb += 4) {
        v2f a, b;
        a.x = As[arow + kb + koff];
        a.y = As[arow + kb + koff + 1];
        b.x = Bs[(kb + koff) * 65 + col];
        b.y = Bs[(kb + koff + 1) * 65 + col];
        acc = wmma4_f32(a, b, acc);
    }
    conv_store_stats(acc, y, pS, pQ, e0, mt, nt, lhalf, l16, tid, blk);
}

// ---------------------------------------------------------------------------
// conv2/conv4: A = lrelu(affine(yprev)), Cin = 64. st = [s[64], t[64]].
// ---------------------------------------------------------------------------
__global__ void __launch_bounds__(256) conv_prev_kernel(
    const float* __restrict__ yprev, const float* __restrict__ st,
    const float* __restrict__ W, float* __restrict__ y,
    float* __restrict__ pS, float* __restrict__ pQ)
{
    __shared__ float As[32 * 65];
    __shared__ float Bs[64 * 65];
    const int tid = threadIdx.x, lane = tid & 31, wave = tid >> 5;
    const int mt = wave >> 2, nt = wave & 3, lhalf = lane >> 4, l16 = lane & 15;
    const int blk = blockIdx.x, e0 = blk * 32;

    // stage A: coalesced read of yprev rows, affine + lrelu applied here
    for (int e = tid; e < 32 * 64; e += 256) {
        int r = e >> 6, c = e & 63;
        float v = yprev[(size_t)(e0 + r) * 64 + c] * st[c] + st[64 + c];
        As[r * 65 + c] = lrelu(v);
    }
    for (int e = tid; e < 64 * 64; e += 256) {
        int o = e >> 6, c = e & 63;
        Bs[c * 65 + o] = W[o * 64 + c];
    }
    __syncthreads();

    const int arow = (mt * 16 + l16) * 65;
    const int col = nt * 16 + l16;
    const int koff = lhalf * 2;
    v8f acc = {};
#pragma unroll
    for (int kb = 0; kb < 64; kb += 4) {
        v2f a, b;
        a.x = As[arow + kb + koff];
        a.y = As[arow + kb + koff + 1];
        b.x = Bs[(kb + koff) * 65 + col];
        b.y = Bs[(kb + koff + 1) * 65 + col];
        acc = wmma4_f32(a, b, acc);
    }
    conv_store_stats(acc, y, pS, pQ, e0, mt, nt, lhalf, l16, tid, blk);
}

// ---------------------------------------------------------------------------
// Fold per-block partials into BN affine: s = g*rsqrt(var+eps), t = b - mean*s.
// One block per channel, deterministic tree reduce.
// ---------------------------------------------------------------------------
__global__ void __launch_bounds__(256) affine_kernel(
    const float* __restrict__ pS, const float* __restrict__ pQ, int nblk, float invM,
    const float* __restrict__ g, const float* __restrict__ bta, float* __restrict__ stOut)
{
    __shared__ float sS[256];
    __shared__ float sQ[256];
    const int ch = blockIdx.x;
    const int t = threadIdx.x;
    float s = 0.f, q = 0.f;
    for (int i = t; i < nblk; i += 256) {
        s += pS[(size_t)i * 64 + ch];
        q += pQ[(size_t)i * 64 + ch];
    }
    sS[t] = s; sQ[t] = q;
    __syncthreads();
    for (int k = 128; k > 0; k >>= 1) {
        if (t < k) { sS[t] += sS[t + k]; sQ[t] += sQ[t + k]; }
        __syncthreads();
    }
    if (t == 0) {
        float m = sS[0] * invM;
        float v = sQ[0] * invM - m * m;
        float sc = g[ch] * rsqrtf(v + BN_EPS);
        stOut[ch] = sc;
        stOut[64 + ch] = bta[ch] - m * sc;
    }
}

// ---------------------------------------------------------------------------
// maxpool over k neighbors of lrelu(affine(y)): xout[n][c]
// ---------------------------------------------------------------------------
__global__ void __launch_bounds__(256) maxpool_kernel(
    const float* __restrict__ y, const float* __restrict__ st, float* __restrict__ xout)
{
    const int gid = blockIdx.x * 256 + threadIdx.x;   // NPTS*64 threads
    const int n = gid >> 6, c = gid & 63;
    const float s = st[c], t = st[64 + c];
    float m = -__builtin_inff();
    for (int kk = 0; kk < KNBR; ++kk) {
        float v = y[(size_t)(n * KNBR + kk) * 64 + c] * s + t;
        v = lrelu(v);
        m = fmaxf(m, v);
    }
    xout[(size_t)n * 64 + c] = m;
}

// ---------------------------------------------------------------------------
// MLP layer 1: h = relu(concat(x1,x2,x3) @ W9a^T + b9a), M=8192 K=192 N=128.
// grid = (NPTS/32, 2); block = 32 rows x 64 cols. K split in 2 phases of 96
// so As (24.7KB) + Bs (25.0KB) fit in LDS.
// ---------------------------------------------------------------------------
__global__ void __launch_bounds__(256) mlp1_kernel(
    const float* __restrict__ x1, const float* __restrict__ x2, const float* __restrict__ x3,
    const float* __restrict__ W, const float* __restrict__ bias, float* __restrict__ h)
{
    __shared__ float As[32 * 193];
    __shared__ float Bs[96 * 65];
    const int tid = threadIdx.x, lane = tid & 31, wave = tid >> 5;
    const int mt = wave >> 2, nt = wave & 3, lhalf = lane >> 4, l16 = lane & 15;
    const int n0 = blockIdx.x * 32;
    const int cb = blockIdx.y;

    // stage A rows = concat(x1,x2,x3)
    {
        const int tr = tid >> 3;
        const int rn = n0 + tr;
        for (int c = (tid & 7); c < 192; c += 8) {
            float f = (c < 64) ? x1[(size_t)rn * 64 + c]
                               : ((c < 128) ? x2[(size_t)rn * 64 + c - 64]
                                            : x3[(size_t)rn * 64 + c - 128]);
            As[tr * 193 + c] = f;
        }
    }

    const int arow = (mt * 16 + l16) * 193;
    const int col = nt * 16 + l16;                 // 0..63 local
    const int gcol = cb * 64 + col;                // global out channel
    const int koff = lhalf * 2;
    v8f acc = {};
    for (int p = 0; p < 2; ++p) {
        __syncthreads();                           // As ready (p=0) / Bs reuse safe (p=1)
        for (int e = tid; e < 64 * 96; e += 256) {
            int o = e / 96, c = e % 96;
            Bs[c * 65 + o] = W[(size_t)(cb * 64 + o) * 192 + p * 96 + c];
        }
        __syncthreads();
#pragma unroll
        for (int kb = 0; kb < 96; kb += 4) {
            v2f a, b;
            a.x = As[arow + p * 96 + kb + koff];
            a.y = As[arow + p * 96 + kb + koff + 1];
            b.x = Bs[(kb + koff) * 65 + col];
            b.y = Bs[(kb + koff + 1) * 65 + col];
            acc = wmma4_f32(a, b, acc);
        }
    }
    const float bb = bias[gcol];
#pragma unroll
    for (int v = 0; v < 8; ++v) {
        int r = n0 + mt * 16 + lhalf * 8 + v;
        float val = acc[v] + bb;
        h[(size_t)r * 128 + gcol] = val > 0.f ? val : 0.f;
    }
}

// ---------------------------------------------------------------------------
// MLP layer 2 (tiny): out[o][n] = h[n] . W9b[o] + b9b[o]; output (16, 8192)
// ---------------------------------------------------------------------------
__global__ void __launch_bounds__(256) mlp2_kernel(
    const float* __restrict__ h, const float* __restrict__ W,
    const float* __restrict__ bias, float* __restrict__ out)
{
    const int gid = blockIdx.x * 256 + threadIdx.x;   // NPTS*16 threads
    const int n = gid >> 4, o = gid & 15;
    float acc = bias[o];
    for (int c = 0; c < 128; ++c)
        acc += h[(size_t)n * 128 + c] * W[o * 128 + c];
    out[(size_t)o * NPTS + n] = acc;
}

// ---------------------------------------------------------------------------
extern "C" void kernel_launch(void* const* d_in, const int* in_sizes, int n_in,
                              void* d_out, int out_size, void* d_ws, size_t ws_size,
                              hipStream_t stream)
{
    (void)in_sizes; (void)n_in; (void)out_size; (void)ws_size;
    const float* x   = (const float*)d_in[0];
    /* d_in[1] = k (==20, hardcoded) */
    const float* W1  = (const float*)d_in[2];
    const float* g1  = (const float*)d_in[3];
    const float* b1  = (const float*)d_in[4];
    const float* W2  = (const float*)d_in[5];
    const float* g2  = (const float*)d_in[6];
    const float* b2  = (const float*)d_in[7];
    const float* W3  = (const float*)d_in[8];
    const float* g3  = (const float*)d_in[9];
    const float* b3  = (const float*)d_in[10];
    const float* W4  = (const float*)d_in[11];
    const float* g4  = (const float*)d_in[12];
    const float* b4  = (const float*)d_in[13];
    const float* W5  = (const float*)d_in[14];
    const float* g5  = (const float*)d_in[15];
    const float* b5  = (const float*)d_in[16];
    const float* W9a = (const float*)d_in[17];
    const float* b9a = (const float*)d_in[18];
    const float* W9b = (const float*)d_in[19];
    const float* b9b = (const float*)d_in[20];
    float* out = (float*)d_out;

    char* w = (char*)d_ws;
    int*   idx = (int*)w;     w += (size_t)MEDGE * 4;            // 0.66 MB
    float* yA  = (float*)w;   w += (size_t)MEDGE * 64 * 4;       // 41.9 MB
    float* yB  = (float*)w;   w += (size_t)MEDGE * 64 * 4;       // 41.9 MB
    float* pS  = (float*)w;   w += (size_t)CONVBLK * 64 * 4;     // 1.3 MB
    float* pQ  = (float*)w;   w += (size_t)CONVBLK * 64 * 4;     // 1.3 MB
    float* st  = (float*)w;   w += 4096;                         // 5 x 128 floats
    float* x1  = (float*)w;   w += (size_t)NPTS * 64 * 4;        // 2.1 MB
    float* x2  = (float*)w;   w += (size_t)NPTS * 64 * 4;
    float* x3  = (float*)w;   w += (size_t)NPTS * 64 * 4;
    float* hb  = (float*)w;   w += (size_t)NPTS * 128 * 4;       // 4.2 MB

    const float invM = 1.f / (float)MEDGE;

    knn_kernel<<<NPTS, 256, 0, stream>>>(x, idx);

    // edge-conv block 1
    conv1_kernel<<<CONVBLK, 256, 0, stream>>>(x, idx, W1, yA, pS, pQ);
    affine_kernel<<<64, 256, 0, stream>>>(pS, pQ, CONVBLK, invM, g1, b1, st + 0 * 128);
    conv_prev_kernel<<<CONVBLK, 256, 0, stream>>>(yA, st + 0 * 128, W2, yB, pS, pQ);
    affine_kernel<<<64, 256, 0, stream>>>(pS, pQ, CONVBLK, invM, g2, b2, st + 1 * 128);
    maxpool_kernel<<<(NPTS * 64) / 256, 256, 0, stream>>>(yB, st + 1 * 128, x1);

    // edge-conv block 2
    conv_edge64_kernel<<<CONVBLK, 256, 0, stream>>>(x1, idx, W3, yA, pS, pQ);
    affine_kernel<<<64, 256, 0, stream>>>(pS, pQ, CONVBLK, invM, g3, b3, st + 2 * 128);
    conv_prev_kernel<<<CONVBLK, 256, 0, stream>>>(yA, st + 2 * 128, W4, yB, pS, pQ);
    affine_kernel<<<64, 256, 0, stream>>>(pS, pQ, CONVBLK, invM, g4, b4, st + 3 * 128);
    maxpool_kernel<<<(NPTS * 64) / 256, 256, 0, stream>>>(yB, st + 3 * 128, x2);

    // edge-conv block 3
    conv_edge64_kernel<<<CONVBLK, 256, 0, stream>>>(x2, idx, W5, yA, pS, pQ);
    affine_kernel<<<64, 256, 0, stream>>>(pS, pQ, CONVBLK, invM, g5, b5, st + 4 * 128);
    maxpool_kernel<<<(NPTS * 64) / 256, 256, 0, stream>>>(yA, st + 4 * 128, x3);

    // final MLP
    mlp1_kernel<<<dim3(NPTS / 32, 2), 256, 0, stream>>>(x1, x2, x3, W9a, b9a, hb);
    mlp2_kernel<<<(NPTS * 16) / 256, 256, 0, stream>>>(hb, W9b, b9b, out);
}